// Hessian7_67070209295165
// MI455X (gfx1250) — compile-verified
//
#include <hip/hip_runtime.h>
#include <hip/hip_bf16.h>
#include <math.h>
#include <stdint.h>

typedef float v2f __attribute__((ext_vector_type(2)));
typedef float v8f __attribute__((ext_vector_type(8)));

__device__ __forceinline__ float silu(float x) {
  return x / (1.0f + __expf(-x));
}

__device__ __forceinline__ v8f wmma4(v2f a, v2f b, v8f c) {
  // D(16x16) = A(16x4) * B(4x16) + C, fp32
  return __builtin_amdgcn_wmma_f32_16x16x4_f32(false, a, false, b, (short)0, c,
                                               false, false);
}

__device__ __forceinline__ v8f bias8(float b) {
  v8f c;
  #pragma unroll
  for (int r = 0; r < 8; r++) c[r] = b;
  return c;
}

// Async global -> LDS copy (CDNA5 GLOBAL_LOAD_ASYNC_TO_LDS_B32, ASYNCcnt).
// lds_addr is the per-lane LDS byte address (low 32 bits of the generic
// pointer), gaddr the 64-bit global address.
__device__ __forceinline__ void async_g2l_b32(unsigned lds_addr,
                                              const float* gaddr) {
  asm volatile("global_load_async_to_lds_b32 %0, %1, off"
               :: "v"(lds_addr), "v"((unsigned long long)(uintptr_t)gaddr)
               : "memory");
}

__device__ __forceinline__ void wait_async0() {
  asm volatile("s_wait_asynccnt 0x0" ::: "memory");
}

// Generic 16xK @ Kx16 tile GEMM, A and B both resident in LDS, zero-padded.
// A: row-major 16 x ASTR (row = M), B: row-major K x BSTR (col = N), result
// accumulated into C (D-fragment layout). KC = number of K=4 chunks.
template <int KC, int ASTR, int BSTR>
__device__ __forceinline__ v8f gemm_lds(const float* __restrict__ A,
                                        const float* __restrict__ B,
                                        int n0, v8f c, int l16, int half) {
  #pragma unroll
  for (int k0 = 0; k0 < KC * 4; k0 += 4) {
    const int kk = k0 + 2 * half;          // A layout: lane half selects K pair
    v2f a;
    a.x = A[l16 * ASTR + kk];
    a.y = A[l16 * ASTR + kk + 1];
    v2f b;
    b.x = B[kk * BSTR + n0 + l16];
    b.y = B[(kk + 1) * BSTR + n0 + l16];
    c = wmma4(a, b, c);
  }
  return c;
}

// ---------------------------------------------------------------------------
// Gated block, WMMA version. 64 threads = 2 waves, 16 atoms per wave.
//   vmix = v(3x30) @ Wmix(30x60)  -> vV cols 0..29 (norms), vW cols 30..59
//   x    = dense2(concat(s,|vV|), Wa(60x30), ba, Wb(30x60), bb)
//   s_out = silu(x[:,0:30]); v_out = x[:,30:60] * vW
// All weight panels zero-padded in LDS so fragment loads are unconditional.
// ---------------------------------------------------------------------------
__global__ void __launch_bounds__(64)
gated_block_wmma(const float* __restrict__ s_in, const float* __restrict__ v_in,
                 const float* __restrict__ Wmix,
                 const float* __restrict__ Wa, const float* __restrict__ ba,
                 const float* __restrict__ Wb, const float* __restrict__ bb,
                 float* __restrict__ s_out, float* __restrict__ v_out, int N) {
  __shared__ float WmV[32 * 32];   // Wmix cols  0..29, padded 32x32
  __shared__ float WmW[32 * 32];   // Wmix cols 30..59, padded 32x32
  __shared__ float WaL[64 * 32];   // Wa padded 64x32
  __shared__ float WbS[32 * 32];   // Wb cols  0..29, padded
  __shared__ float WbG[32 * 32];   // Wb cols 30..59, padded
  __shared__ float baL[32], bbS[32], bbG[32];
  __shared__ float AV[2][3][16 * 32];   // per-wave v panels (K padded to 32)
  __shared__ float AS[2][16 * 64];      // per-wave concat(s,|vV|) panel (K pad 64)
  __shared__ float HH[2][16 * 32];      // per-wave hidden panel

  const int tid = threadIdx.x;
  const int lane = tid & 31, wave = tid >> 5;
  const int half = lane >> 4, l16 = lane & 15;

  // ---- cooperative zero-padded weight staging ----
  for (int idx = tid; idx < 32 * 32; idx += 64) {
    const int r = idx >> 5, c = idx & 31;
    const bool ok = (r < 30) && (c < 30);
    WmV[idx] = ok ? Wmix[r * 60 + c] : 0.f;
    WmW[idx] = ok ? Wmix[r * 60 + 30 + c] : 0.f;
    WbS[idx] = ok ? Wb[r * 60 + c] : 0.f;
    WbG[idx] = ok ? Wb[r * 60 + 30 + c] : 0.f;
  }
  for (int idx = tid; idx < 64 * 32; idx += 64) {
    const int r = idx >> 5, c = idx & 31;
    WaL[idx] = (r < 60 && c < 30) ? Wa[r * 30 + c] : 0.f;
  }
  if (tid < 32) {
    baL[tid] = (tid < 30) ? ba[tid] : 0.f;
    bbS[tid] = (tid < 30) ? bb[tid] : 0.f;
    bbG[tid] = (tid < 30) ? bb[30 + tid] : 0.f;
  }

  // ---- per-wave activation staging (16 atoms) ----
  const int n0 = blockIdx.x * 32 + wave * 16;
  float* av = &AV[wave][0][0];
  for (int idx = lane; idx < 3 * 16 * 32; idx += 32) {
    const int d = idx >> 9, rem = idx & 511, m = rem >> 5, k = rem & 31;
    const int nm = min(n0 + m, N - 1);
    av[idx] = (k < 30) ? v_in[(size_t)nm * 90 + d * 30 + k] : 0.f;
  }
  float* as_ = &AS[wave][0];
  for (int idx = lane; idx < 16 * 64; idx += 32) {
    const int m = idx >> 6, k = idx & 63;
    const int nm = min(n0 + m, N - 1);
    as_[idx] = (k < 30) ? s_in[(size_t)nm * 30 + k] : 0.f;
  }
  __syncthreads();

  // ---- vmix V-half: 3 directions, 2 N-tiles; |vV| -> AS cols 30..61 ----
  #pragma unroll
  for (int nt = 0; nt < 2; nt++) {
    v8f c0 = gemm_lds<8, 32, 32>(&AV[wave][0][0], WmV, nt * 16, bias8(0.f), l16, half);
    v8f c1 = gemm_lds<8, 32, 32>(&AV[wave][1][0], WmV, nt * 16, bias8(0.f), l16, half);
    v8f c2 = gemm_lds<8, 32, 32>(&AV[wave][2][0], WmV, nt * 16, bias8(0.f), l16, half);
    #pragma unroll
    for (int r = 0; r < 8; r++) {
      const int row = half ? (r + 8) : r;
      const float nv = sqrtf(c0[r] * c0[r] + c1[r] * c1[r] + c2[r] * c2[r]);
      as_[row * 64 + 30 + nt * 16 + l16] = nv;   // cols 60/61 get exact zeros
    }
  }

  // ---- vmix W-half: keep D-fragments live for the gating product ----
  v8f cw[3][2];
  #pragma unroll
  for (int d = 0; d < 3; d++)
    #pragma unroll
    for (int nt = 0; nt < 2; nt++)
      cw[d][nt] = gemm_lds<8, 32, 32>(&AV[wave][d][0], WmW, nt * 16, bias8(0.f), l16, half);

  // ---- dense1: (16x64) @ WaL(64x32), silu -> HH ----
  float* hh = &HH[wave][0];
  #pragma unroll
  for (int nt = 0; nt < 2; nt++) {
    v8f c = gemm_lds<16, 64, 32>(as_, WaL, nt * 16, bias8(baL[nt * 16 + l16]), l16, half);
    #pragma unroll
    for (int r = 0; r < 8; r++) {
      const int row = half ? (r + 8) : r;
      hh[row * 32 + nt * 16 + l16] = silu(c[r]);
    }
  }

  // ---- dense2 s-half: silu -> s_out ----
  #pragma unroll
  for (int nt = 0; nt < 2; nt++) {
    v8f c = gemm_lds<8, 32, 32>(hh, WbS, nt * 16, bias8(bbS[nt * 16 + l16]), l16, half);
    const int o = nt * 16 + l16;
    if (o < 30) {
      #pragma unroll
      for (int r = 0; r < 8; r++) {
        const int row = half ? (r + 8) : r;
        s_out[(size_t)(n0 + row) * 30 + o] = silu(c[r]);
      }
    }
  }

  // ---- dense2 gate-half: v_out = gate * vW (both in D-fragment layout) ----
  #pragma unroll
  for (int nt = 0; nt < 2; nt++) {
    v8f g = gemm_lds<8, 32, 32>(hh, WbG, nt * 16, bias8(bbG[nt * 16 + l16]), l16, half);
    const int o = nt * 16 + l16;
    if (o < 30) {
      #pragma unroll
      for (int d = 0; d < 3; d++) {
        #pragma unroll
        for (int r = 0; r < 8; r++) {
          const int row = half ? (r + 8) : r;
          v_out[(size_t)(n0 + row) * 90 + d * 30 + o] = g[r] * cw[d][nt][r];
        }
      }
    }
  }
}

// ---------------------------------------------------------------------------
// pos MLP + feat MLP, WMMA version. 64 threads = 2 waves, 16 atoms per wave.
//   pos  = dense2([R|onehot(Z)](16x6), Wp1(6x30), Wp2(30x30))
//   feat = dense2([s|vflat|pos](16x150), Wf1(150x30), Wf2(30x10)) -> f (16x10)
// ---------------------------------------------------------------------------
__global__ void __launch_bounds__(64)
posfeat_wmma(const int* __restrict__ Z, const float* __restrict__ R,
             const float* __restrict__ s2, const float* __restrict__ v2,
             const float* __restrict__ Wp1, const float* __restrict__ bp1,
             const float* __restrict__ Wp2, const float* __restrict__ bp2,
             const float* __restrict__ Wf1, const float* __restrict__ bf1,
             const float* __restrict__ Wf2, const float* __restrict__ bf2,
             float* __restrict__ f, int N) {
  __shared__ float Wp1L[8 * 32];     // 6x30  -> 8x32
  __shared__ float Wp2L[32 * 32];    // 30x30 -> 32x32
  __shared__ float Wf1L[160 * 32];   // 150x30-> 160x32
  __shared__ float Wf2L[32 * 16];    // 30x10 -> 32x16
  __shared__ float bp1L[32], bp2L[32], bf1L[32], bf2L[16];
  __shared__ float A6[2][16 * 8];    // per-wave [R|onehot] panel
  __shared__ float AF[2][16 * 160];  // per-wave feat-input panel
  __shared__ float HH[2][16 * 32];   // per-wave hidden panel

  const int tid = threadIdx.x;
  const int lane = tid & 31, wave = tid >> 5;
  const int half = lane >> 4, l16 = lane & 15;

  for (int idx = tid; idx < 8 * 32; idx += 64) {
    const int r = idx >> 5, c = idx & 31;
    Wp1L[idx] = (r < 6 && c < 30) ? Wp1[r * 30 + c] : 0.f;
  }
  for (int idx = tid; idx < 32 * 32; idx += 64) {
    const int r = idx >> 5, c = idx & 31;
    Wp2L[idx] = (r < 30 && c < 30) ? Wp2[r * 30 + c] : 0.f;
  }
  for (int idx = tid; idx < 160 * 32; idx += 64) {
    const int r = idx >> 5, c = idx & 31;
    Wf1L[idx] = (r < 150 && c < 30) ? Wf1[r * 30 + c] : 0.f;
  }
  for (int idx = tid; idx < 32 * 16; idx += 64) {
    const int r = idx >> 4, c = idx & 15;
    Wf2L[idx] = (r < 30 && c < 10) ? Wf2[r * 10 + c] : 0.f;
  }
  if (tid < 32) {
    bp1L[tid] = (tid < 30) ? bp1[tid] : 0.f;
    bp2L[tid] = (tid < 30) ? bp2[tid] : 0.f;
    bf1L[tid] = (tid < 30) ? bf1[tid] : 0.f;
  }
  if (tid < 16) bf2L[tid] = (tid < 10) ? bf2[tid] : 0.f;

  const int n0 = blockIdx.x * 32 + wave * 16;
  float* a6 = &A6[wave][0];
  for (int idx = lane; idx < 16 * 8; idx += 32) {
    const int m = idx >> 3, k = idx & 7;
    const int nm = min(n0 + m, N - 1);
    float x = 0.f;
    if (k < 3) {
      x = R[(size_t)nm * 3 + k];
    } else if (k < 6) {
      const int z = Z[nm];
      const int want = (k == 3) ? 1 : (k == 4) ? 6 : 8;
      x = (z == want) ? 1.f : 0.f;
    }
    a6[idx] = x;
  }
  float* af = &AF[wave][0];
  for (int idx = lane; idx < 16 * 160; idx += 32) {
    const int m = idx / 160, c = idx % 160;
    const int nm = min(n0 + m, N - 1);
    float x = 0.f;
    if (c < 30)       x = s2[(size_t)nm * 30 + c];
    else if (c < 120) x = v2[(size_t)nm * 90 + (c - 30)];
    af[idx] = x;   // cols 120..151 filled by pos below; 152..159 stay zero
  }
  __syncthreads();

  float* hh = &HH[wave][0];

  // pos1: K=8 (2 chunks), silu -> HH
  #pragma unroll
  for (int nt = 0; nt < 2; nt++) {
    v8f c = gemm_lds<2, 8, 32>(a6, Wp1L, nt * 16, bias8(bp1L[nt * 16 + l16]), l16, half);
    #pragma unroll
    for (int r = 0; r < 8; r++) {
      const int row = half ? (r + 8) : r;
      hh[row * 32 + nt * 16 + l16] = silu(c[r]);
    }
  }
  // pos2: K=32 (8 chunks), linear; write into AF cols 120..151 (150/151 exact 0)
  #pragma unroll
  for (int nt = 0; nt < 2; nt++) {
    v8f c = gemm_lds<8, 32, 32>(hh, Wp2L, nt * 16, bias8(bp2L[nt * 16 + l16]), l16, half);
    #pragma unroll
    for (int r = 0; r < 8; r++) {
      const int row = half ? (r + 8) : r;
      af[row * 160 + 120 + nt * 16 + l16] = c[r];
    }
  }
  // feat1: K=160 (40 chunks), silu -> HH
  #pragma unroll
  for (int nt = 0; nt < 2; nt++) {
    v8f c = gemm_lds<40, 160, 32>(af, Wf1L, nt * 16, bias8(bf1L[nt * 16 + l16]), l16, half);
    #pragma unroll
    for (int r = 0; r < 8; r++) {
      const int row = half ? (r + 8) : r;
      hh[row * 32 + nt * 16 + l16] = silu(c[r]);
    }
  }
  // feat2: K=32 (8 chunks), one 16-col tile (cols >= 10 masked on store)
  {
    v8f c = gemm_lds<8, 32, 16>(hh, Wf2L, 0, bias8(bf2L[l16]), l16, half);
    if (l16 < 10) {
      #pragma unroll
      for (int r = 0; r < 8; r++) {
        const int row = half ? (r + 8) : r;
        f[(size_t)(n0 + row) * 10 + l16] = c[r];
      }
    }
  }
}

// ---------------------------------------------------------------------------
// Pair Hessian stage, WMMA. 128 threads = 4 waves = 4 j-tiles per (b, i).
//   X(16x20) = [f_i | f_j];  H = silu(X@Wg1+bg1);  O = H@Wg2+bg2
//   scatter O into hess[b, i*3+p, j*3+q]
// f_i (40 B) and the 16 f_j rows (640 B, contiguous) are brought into LDS via
// CDNA5 async global->LDS copies (ASYNCcnt + s_wait_asynccnt).
// ---------------------------------------------------------------------------
__global__ void __launch_bounds__(128)
pair_hessian_kernel(const float* __restrict__ f,
                    const float* __restrict__ Wg1, const float* __restrict__ bg1,
                    const float* __restrict__ Wg2, const float* __restrict__ bg2,
                    float* __restrict__ out, int A, int Bm) {
  __shared__ float Wg1L[20 * 32];   // 20x30 -> 20x32
  __shared__ float Wg2L[32 * 16];   // 30x9  -> 32x16
  __shared__ float bg1L[32], bg2L[16];
  __shared__ float FI[4][16];       // per-wave f_i (10 used)
  __shared__ float FJ[4][16 * 10];  // per-wave f_j rows
  __shared__ float HH[4][16 * 32];  // per-wave hidden panel

  const int tid = threadIdx.x;
  const int lane = tid & 31, wave = tid >> 5;
  const int half = lane >> 4, l16 = lane & 15;

  for (int idx = tid; idx < 20 * 32; idx += 128) {
    const int r = idx >> 5, c = idx & 31;
    Wg1L[idx] = (c < 30) ? Wg1[r * 30 + c] : 0.f;
  }
  for (int idx = tid; idx < 32 * 16; idx += 128) {
    const int r = idx >> 4, c = idx & 15;
    Wg2L[idx] = (r < 30 && c < 9) ? Wg2[r * 9 + c] : 0.f;
  }
  if (tid < 32) bg1L[tid] = (tid < 30) ? bg1[tid] : 0.f;
  if (tid < 16) bg2L[tid] = (tid < 9) ? bg2[tid] : 0.f;

  const int i = blockIdx.x % A;   // anchor atom
  const int b = blockIdx.x / A;   // molecule
  const int j0 = wave * 16;       // j-tile (4 waves cover A=60)
  const float* fi = f + ((size_t)b * A + i) * 10;

  // ---- async global->LDS staging of f_i and valid f_j rows ----
  {
    const int nvalid = min(16, A - j0);                 // rows in this j-tile
    const float* fjsrc = f + ((size_t)b * A + j0) * 10; // contiguous rows
    const unsigned ldsFI = (unsigned)(uintptr_t)&FI[wave][0];
    const unsigned ldsFJ = (unsigned)(uintptr_t)&FJ[wave][0];
    if (lane < 10) async_g2l_b32(ldsFI + lane * 4, fi + lane);
    for (int t = lane; t < nvalid * 10; t += 32)
      async_g2l_b32(ldsFJ + t * 4, fjsrc + t);
    wait_async0();
  }
  __syncthreads();

  // GEMM1: K=20 (5 chunks), 2 N-tiles; A-fragments select FI/FJ regions
  const float* fiL = &FI[wave][0];
  const float* fjL = &FJ[wave][0];
  v8f c0 = bias8(bg1L[l16]);
  v8f c1 = bias8(bg1L[16 + l16]);
  #pragma unroll
  for (int k0 = 0; k0 < 20; k0 += 4) {
    const int kk = k0 + 2 * half;
    v2f a;
    a.x = (kk     < 10) ? fiL[kk]     : fjL[l16 * 10 + kk - 10];
    a.y = (kk + 1 < 10) ? fiL[kk + 1] : fjL[l16 * 10 + kk + 1 - 10];
    v2f b0, b1;
    b0.x = Wg1L[kk * 32 + l16];
    b0.y = Wg1L[(kk + 1) * 32 + l16];
    b1.x = Wg1L[kk * 32 + 16 + l16];
    b1.y = Wg1L[(kk + 1) * 32 + 16 + l16];
    c0 = wmma4(a, b0, c0);
    c1 = wmma4(a, b1, c1);
  }

  // silu -> HH (cols >= 30 are exact zeros via padded weights/bias)
  float* hl = &HH[wave][0];
  #pragma unroll
  for (int r = 0; r < 8; r++) {
    const int row = half ? (r + 8) : r;
    hl[row * 32 + l16]      = silu(c0[r]);
    hl[row * 32 + 16 + l16] = silu(c1[r]);
  }

  // GEMM2: K=32 (8 chunks), one 16-col tile (cols >= 9 dead)
  v8f c2 = gemm_lds<8, 32, 16>(hl, Wg2L, 0, bias8(bg2L[l16]), l16, half);

  // scatter: out[b, i*3+p, j*3+q]
  if (l16 < 9 && b < Bm) {
    const int p = l16 / 3, q = l16 % 3;
    const int A3 = 3 * A;
    #pragma unroll
    for (int r = 0; r < 8; r++) {
      const int M = half ? (r + 8) : r;
      const int j = j0 + M;
      if (j < A)
        out[(size_t)b * A3 * A3 + (size_t)(i * 3 + p) * A3 + (j * 3 + q)] = c2[r];
    }
  }
}

// ---------------------------------------------------------------------------
extern "C" void kernel_launch(void* const* d_in, const int* in_sizes, int n_in,
                              void* d_out, int out_size, void* d_ws, size_t ws_size,
                              hipStream_t stream) {
  const int*   Z     = (const int*)  d_in[0];
  const float* R     = (const float*)d_in[1];
  const float* l0    = (const float*)d_in[2];
  const float* l1    = (const float*)d_in[3];
  const float* Wmix1 = (const float*)d_in[4];
  const float* W1a   = (const float*)d_in[5];
  const float* b1a   = (const float*)d_in[6];
  const float* W1b   = (const float*)d_in[7];
  const float* b1b   = (const float*)d_in[8];
  const float* Wmix2 = (const float*)d_in[9];
  const float* W2a   = (const float*)d_in[10];
  const float* b2a   = (const float*)d_in[11];
  const float* W2b   = (const float*)d_in[12];
  const float* b2b   = (const float*)d_in[13];
  const float* Wp1   = (const float*)d_in[14];
  const float* bp1   = (const float*)d_in[15];
  const float* Wp2   = (const float*)d_in[16];
  const float* bp2   = (const float*)d_in[17];
  const float* Wf1   = (const float*)d_in[18];
  const float* bf1   = (const float*)d_in[19];
  const float* Wf2   = (const float*)d_in[20];
  const float* bf2   = (const float*)d_in[21];
  const float* Wg1   = (const float*)d_in[22];
  const float* bg1   = (const float*)d_in[23];
  const float* Wg2   = (const float*)d_in[24];
  const float* bg2   = (const float*)d_in[25];

  const int N  = in_sizes[0];   // 15360
  const int A  = 60;            // fixed in reference module
  const int Bm = N / A;         // 256

  float* ws = (float*)d_ws;
  float* s1 = ws;
  float* v1 = s1 + (size_t)N * 30;
  float* s2 = v1 + (size_t)N * 90;
  float* v2 = s2 + (size_t)N * 30;
  float* fb = v2 + (size_t)N * 90;   // total N*250 floats ~= 15.4 MB

  const int nBlk = (N + 31) / 32;    // 32 atoms (2 waves) per block
  gated_block_wmma<<<dim3(nBlk), dim3(64), 0, stream>>>(l0, l1, Wmix1, W1a, b1a,
                                                        W1b, b1b, s1, v1, N);
  gated_block_wmma<<<dim3(nBlk), dim3(64), 0, stream>>>(s1, v1, Wmix2, W2a, b2a,
                                                        W2b, b2b, s2, v2, N);
  posfeat_wmma<<<dim3(nBlk), dim3(64), 0, stream>>>(Z, R, s2, v2, Wp1, bp1, Wp2,
                                                    bp2, Wf1, bf1, Wf2, bf2, fb, N);
  pair_hessian_kernel<<<dim3(Bm * A), dim3(128), 0, stream>>>(fb, Wg1, bg1, Wg2,
                                                              bg2, (float*)d_out,
                                                              A, Bm);
}